// PostProcessor_60266981097822
// MI455X (gfx1250) — compile-verified
//
#include <hip/hip_runtime.h>
#include <stdint.h>

#define NMS_T   0.5f
#define OFF_PC  2666.0f   /* 2 * 1333 per-class offset */
#define KTOP    100
#define NPAD    8192      /* bitonic sort capacity (power of two); N == 8192 */
#define NWORDS  256       /* N/32 mask words per row */
#define NSLOT   8         /* NWORDS / 32 lanes */

typedef __attribute__((ext_vector_type(2))) float v2f;
typedef __attribute__((ext_vector_type(8))) float v8f;

// ---------------------------------------------------------------------------
// 1) key[i] = iou_scores[i*C + labels[i]] as diag(A x B^T) with one-hot B,
//    computed on the matrix pipe: V_WMMA_F32_16X16X4_F32, f32 in / f32 acc,
//    bit-exact (each product is x*1 or x*0; each acc adds at most one nonzero).
//    One wave handles 16 rows; 21 K-chunks of 4 cover C=81.
// ---------------------------------------------------------------------------
__global__ __launch_bounds__(256) void k_key_wmma(const float* __restrict__ iou_scores,
                                                  const int* __restrict__ labels,
                                                  float* __restrict__ key, int n, int C) {
    const int wave = threadIdx.x >> 5;
    const int lane = threadIdx.x & 31;
    const int m0   = (blockIdx.x * 8 + wave) * 16;   // row-group base
    const int r    = lane & 15;                      // row/col within group
    const int hi   = lane >> 4;                      // lane half -> K pair {0,1} or {2,3}
    const int row  = m0 + r;
    const bool rowv = (row < n);

    const int lbl = rowv ? labels[row] : -1;         // label of column N=r (one-hot B)

    v8f acc = {};
    const int nchunk = (C + 3) >> 2;                 // 21 for C=81
    for (int c = 0; c < nchunk; ++c) {
        const int k0 = 4 * c + 2 * hi;               // this lane's K pair
        // A (16x4 f32): VGPR0 -> K = k0, VGPR1 -> K = k0+1, M = lane%16
        v2f a;
        a.x = (rowv && k0     < C) ? iou_scores[(size_t)row * C + k0]     : 0.0f;
        a.y = (rowv && k0 + 1 < C) ? iou_scores[(size_t)row * C + k0 + 1] : 0.0f;
        // B (4x16 one-hot f32): B[k][j] = (labels[m0+j] == k_abs), N = lane%16
        v2f b;
        b.x = (lbl == k0)     ? 1.0f : 0.0f;
        b.y = (lbl == k0 + 1) ? 1.0f : 0.0f;
        acc = __builtin_amdgcn_wmma_f32_16x16x4_f32(
                  false, a, false, b, (short)0, acc, false, false);
    }
    // Diagonal extraction per C/D layout: reg s holds (M=s, N=lane) lanes 0-15
    // and (M=s+8, N=lane-16) lanes 16-31 -> diag lives on lanes s and s+24.
#pragma unroll
    for (int s = 0; s < 8; ++s) {
        if (lane == s      && m0 + s     < n) key[m0 + s]     = acc[s];
        if (lane == s + 24 && m0 + s + 8 < n) key[m0 + s + 8] = acc[s];
    }
}

// ---------------------------------------------------------------------------
// 2) single-block bitonic sort of (key desc, index asc) -> order[r]
// ---------------------------------------------------------------------------
__global__ __launch_bounds__(1024) void k_sort(const float* __restrict__ key,
                                               int* __restrict__ order, int n) {
    __shared__ float sk[NPAD];
    __shared__ int   sv[NPAD];
    const int tid = threadIdx.x;
    for (int t = tid; t < NPAD; t += 1024) {
        sk[t] = (t < n) ? key[t] : -__builtin_inff();
        sv[t] = t;
    }
    for (unsigned k = 2; k <= NPAD; k <<= 1) {
        for (unsigned j = k >> 1; j > 0; j >>= 1) {
            __syncthreads();
            for (unsigned t = tid; t < NPAD; t += 1024) {
                unsigned p = t ^ j;
                if (p > t) {
                    float ka = sk[t], kb = sk[p];
                    int   va = sv[t], vb = sv[p];
                    // "t's element belongs after p's element" under
                    // (key descending, index ascending on ties)
                    bool a_after = (ka < kb) || (ka == kb && va > vb);
                    bool doswap  = ((t & k) == 0) ? a_after : !a_after;
                    if (doswap) {
                        sk[t] = kb; sk[p] = ka;
                        sv[t] = vb; sv[p] = va;
                    }
                }
            }
        }
    }
    __syncthreads();
    for (int t = tid; t < n; t += 1024) order[t] = sv[t];
}

// ---------------------------------------------------------------------------
// 3) permute by order; build class-offset boxes
// ---------------------------------------------------------------------------
__global__ void k_gather(const float4* __restrict__ boxes,
                         const float* __restrict__ scores,
                         const int* __restrict__ labels,
                         const int* __restrict__ order,
                         float* __restrict__ s_s, int* __restrict__ l_s,
                         float4* __restrict__ b_orig, float4* __restrict__ b_off,
                         int n) {
    int r = blockIdx.x * blockDim.x + threadIdx.x;
    if (r >= n) return;
    int i = order[r];
    float sc = scores[i];
    int   l  = labels[i];
    float4 b = boxes[i];
    s_s[r] = sc;
    l_s[r] = l;
    b_orig[r] = b;
    float o = (float)l * OFF_PC;
    b_off[r] = make_float4(b.x + o, b.y + o, b.z + o, b.w + o);
}

// ---------------------------------------------------------------------------
// 4) suppression bitmask: mask[i][w] bit b set iff j=32w+b > i and IoU > T.
//    Column tile of 256 boxes staged into LDS with the CDNA5 async copy
//    engine (global_load_async_to_lds_b128 + s_wait_asynccnt).
// ---------------------------------------------------------------------------
__global__ __launch_bounds__(256) void k_mask(const float4* __restrict__ b_off,
                                              unsigned* __restrict__ mask, int n) {
    __shared__ float4 tile[256];
    const int tid = threadIdx.x;
    const int i   = blockIdx.y * 256 + tid;   // row
    const int j0  = blockIdx.x * 256;         // column tile base

    // Async global -> LDS: each lane moves one 16B box of the column tile.
    {
        unsigned lds_addr = (unsigned)(uintptr_t)(&tile[tid]);
        unsigned long long gaddr =
            (unsigned long long)(uintptr_t)(b_off + j0 + tid);
        asm volatile("global_load_async_to_lds_b128 %0, %1, off"
                     :: "v"(lds_addr), "v"(gaddr) : "memory");
    }
    float4 bb = b_off[i];                     // row box (overlaps with async)
    float area_i = (bb.z - bb.x) * (bb.w - bb.y);
    asm volatile("s_wait_asynccnt 0" ::: "memory");
    __syncthreads();

    unsigned w[8] = {0u, 0u, 0u, 0u, 0u, 0u, 0u, 0u};
    for (int jj = 0; jj < 256; ++jj) {
        float4 cb = tile[jj];                 // LDS broadcast read
        float lx = fmaxf(bb.x, cb.x), ly = fmaxf(bb.y, cb.y);
        float rx = fminf(bb.z, cb.z), ry = fminf(bb.w, cb.w);
        float iw = fmaxf(rx - lx, 0.0f), ih = fmaxf(ry - ly, 0.0f);
        float inter  = iw * ih;
        float area_j = (cb.z - cb.x) * (cb.w - cb.y);
        float uni    = area_i + area_j - inter;
        // inter / max(uni,1e-9) > T  <=>  inter > T * max(uni,1e-9)
        bool sup = (j0 + jj > i) && (inter > NMS_T * fmaxf(uni, 1e-9f));
        if (sup) w[jj >> 5] |= 1u << (jj & 31);
    }
    const int nwords = n >> 5;
    size_t base = (size_t)i * nwords + (size_t)blockIdx.x * 8;
#pragma unroll
    for (int s = 0; s < 8; ++s) mask[base + s] = w[s];
}

// ---------------------------------------------------------------------------
// 5) serial greedy scan: ONE wave32. Removal state = 8 regs x 32 lanes
//    (word w lives in remv[w/32] on lane w%32). Critical path per kept row:
//    shfl broadcast -> bit test -> 8 parallel L2 loads -> v_or. No barriers,
//    no LDS round-trips. global_prefetch pulls row i+32 ahead of use.
//    Fused masked-score write uses shfl from keep registers (keep word for
//    t = lane + 32*q is word q, uniform across the wave).
// ---------------------------------------------------------------------------
__global__ __launch_bounds__(32) void k_scan(const unsigned* __restrict__ mask,
                                             const float* __restrict__ s_s,
                                             unsigned* __restrict__ keep_out,
                                             float* __restrict__ masked, int n) {
    const int lane = threadIdx.x;
    unsigned remv[NSLOT]  = {0u, 0u, 0u, 0u, 0u, 0u, 0u, 0u};
    unsigned keepr[NSLOT] = {0u, 0u, 0u, 0u, 0u, 0u, 0u, 0u};

#pragma unroll 1
    for (int s = 0; s < NSLOT; ++s) {
#pragma unroll 1
        for (int wl = 0; wl < 32; ++wl) {          // lane owning this word
            unsigned cur = __shfl(remv[s], wl);    // current removal word
#pragma unroll 1
            for (int b = 0; b < 32; ++b) {
                int i = ((s << 5) + wl) * 32 + b;  // row index
                if (i + 32 < n)                    // prefetch row i+32 (1KB)
                    __builtin_prefetch(&mask[(size_t)(i + 32) * NWORDS + lane * 8], 0, 0);
                if (!((cur >> b) & 1u)) {
                    if (lane == wl) keepr[s] |= 1u << b;
                    size_t base = (size_t)i * NWORDS + lane;
#pragma unroll
                    for (int s2 = 0; s2 < NSLOT; ++s2)
                        remv[s2] |= mask[base + (s2 << 5)];
                    cur = __shfl(remv[s], wl);     // rebroadcast updated word
                }
            }
        }
    }
#pragma unroll
    for (int s = 0; s < NSLOT; ++s) {
        keep_out[(s << 5) + lane] = keepr[s];
        for (int wq = 0; wq < 32; ++wq) {
            unsigned word = __shfl(keepr[s], wq);      // keep word (s*32+wq)
            int t = (((s << 5) + wq) << 5) + lane;     // row; bit index == lane
            if (t < n)
                masked[t] = ((word >> lane) & 1u) ? s_s[t] : -__builtin_inff();
        }
    }
}

// ---------------------------------------------------------------------------
// 6) iterative top-K argmax + write all outputs
//    d_out layout: [0..4K) boxes, [4K..5K) scores, [5K..6K) labels, [6K..7K) valid
// ---------------------------------------------------------------------------
__global__ __launch_bounds__(256) void k_topk(const float* __restrict__ masked,
                                              const float4* __restrict__ b_orig,
                                              const int* __restrict__ l_s,
                                              float* __restrict__ out, int n) {
    __shared__ float sc[NPAD];
    __shared__ float rv[256];
    __shared__ int   ri[256];
    const int tid = threadIdx.x;
    for (int t = tid; t < n; t += 256) sc[t] = masked[t];
    __syncthreads();
    for (int k = 0; k < KTOP; ++k) {
        float bv = -__builtin_inff();
        int   bi = n;
        for (int t = tid; t < n; t += 256) {
            float v = sc[t];
            if (v > bv || (v == bv && t < bi)) { bv = v; bi = t; }
        }
        rv[tid] = bv; ri[tid] = bi;
        __syncthreads();
        for (int s = 128; s > 0; s >>= 1) {
            if (tid < s) {
                float ov = rv[tid + s]; int oi = ri[tid + s];
                if (ov > rv[tid] || (ov == rv[tid] && oi < ri[tid])) {
                    rv[tid] = ov; ri[tid] = oi;
                }
            }
            __syncthreads();
        }
        if (tid == 0) {
            float v = rv[0]; int id = ri[0];
            bool valid = (v != -__builtin_inff());
            float4 b = valid ? b_orig[id] : make_float4(0.f, 0.f, 0.f, 0.f);
            out[k * 4 + 0] = b.x;
            out[k * 4 + 1] = b.y;
            out[k * 4 + 2] = b.z;
            out[k * 4 + 3] = b.w;
            out[4 * KTOP + k] = valid ? v : 0.0f;
            out[5 * KTOP + k] = valid ? (float)l_s[id] : 0.0f;
            out[6 * KTOP + k] = valid ? 1.0f : 0.0f;
            sc[id] = -__builtin_inff();
        }
        __syncthreads();
    }
}

// ---------------------------------------------------------------------------
extern "C" void kernel_launch(void* const* d_in, const int* in_sizes, int n_in,
                              void* d_out, int out_size, void* d_ws, size_t ws_size,
                              hipStream_t stream) {
    const float* boxes      = (const float*)d_in[0];   // [N,4]
    const float* scores     = (const float*)d_in[1];   // [N]
    const float* iou_scores = (const float*)d_in[2];   // [N,C]
    const int*   labels     = (const int*)d_in[3];     // [N]
    const int n = in_sizes[1];
    const int C = in_sizes[2] / n;
    const int nwords = n >> 5;

    // workspace layout (base is 256B aligned):
    float*    key    = (float*)d_ws;                 // n
    int*      order  = (int*)(key + n);              // n
    float*    s_s    = (float*)(order + n);          // n
    int*      l_s    = (int*)(s_s + n);              // n
    float4*   b_orig = (float4*)(l_s + n);           // n (offset 16n bytes, aligned)
    float4*   b_off  = b_orig + n;                   // n
    float*    masked = (float*)(b_off + n);          // n
    unsigned* keepw  = (unsigned*)(masked + n);      // nwords
    unsigned* mask   = keepw + nwords;               // n * nwords  (8 MB @ n=8192)

    const int nblk = (n + 255) / 256;

    k_key_wmma<<<(n + 127) / 128, 256, 0, stream>>>(iou_scores, labels, key, n, C);
    k_sort  <<<1, 1024, 0, stream>>>(key, order, n);
    k_gather<<<nblk, 256, 0, stream>>>((const float4*)boxes, scores, labels,
                                       order, s_s, l_s, b_orig, b_off, n);
    k_mask  <<<dim3(nblk, nblk), 256, 0, stream>>>(b_off, mask, n);
    k_scan  <<<1, 32, 0, stream>>>(mask, s_s, keepw, masked, n);
    k_topk  <<<1, 256, 0, stream>>>(masked, b_orig, l_s, (float*)d_out, n);
}